// SurfaceLoss_76012331205094
// MI455X (gfx1250) — compile-verified
//
#include <hip/hip_runtime.h>

#define VOL   262144      // 64^3
#define NVOX  524288      // B*Z*Y*X = 2*64^3
#define INFV  1000000000.0f

typedef __attribute__((ext_vector_type(2))) float v2f;
typedef __attribute__((ext_vector_type(8))) float v8f;

// Async global->LDS path (gfx1250 ASYNCcnt ops), guarded so absence of the
// builtin falls back to the proven synchronous tile fill.
#if defined(__has_builtin)
#if __has_builtin(__builtin_amdgcn_global_load_async_to_lds_b32) && \
    __has_builtin(__builtin_amdgcn_s_wait_asynccnt)
#define HAVE_ASYNC_LDS 1
#endif
#endif
#ifndef HAVE_ASYNC_LDS
#define HAVE_ASYNC_LDS 0
#endif

#if HAVE_ASYNC_LDS
typedef __attribute__((address_space(1))) int GInt;   // global (AS1) int*
typedef __attribute__((address_space(3))) int LInt;   // LDS    (AS3) int*
#endif

// ---------------------------------------------------------------------------
// Wave-level sum via matrix unit: A holds one partial per lane (a.x), B = ones.
// D[i][j] = rowsum_i, so per-lane s = sum of 8 D regs = sum(rows 0..7) for
// lanes 0..15 and sum(rows 8..15) for lanes 16..31; one xor-16 shuffle closes it.
// Requires EXEC all-ones (call only from uniform control flow).
// ---------------------------------------------------------------------------
__device__ __forceinline__ float wave_sum_wmma(float acc) {
    v2f a;  a.x = acc;  a.y = 0.0f;
    v2f bo; bo.x = 1.0f; bo.y = 1.0f;
    v8f c = {};
    c = __builtin_amdgcn_wmma_f32_16x16x4_f32(
            /*neg_a=*/false, a, /*neg_b=*/false, bo,
            /*c_mod=*/(short)0, c, /*reuse_a=*/false, /*reuse_b=*/false);
    float s = c[0] + c[1] + c[2] + c[3] + c[4] + c[5] + c[6] + c[7];
    s += __shfl_xor(s, 16, 32);
    return s;
}

// ---------------------------------------------------------------------------
// Seed init: f_out = fg ? 0 : INF, f_in = fg ? INF : 0, any_fg |= fg
// vol index: buf(0=fg-seeded,1=bg-seeded)*6 + b*3 + c
// ---------------------------------------------------------------------------
__global__ void k_init(const int* __restrict__ tgt, float* __restrict__ f,
                       int* __restrict__ anyfg) {
    int s = blockIdx.x * 256 + threadIdx.x;      // exactly NVOX threads
    int b = s >> 18;
    int r = s & (VOL - 1);
    int t = tgt[s];
#pragma unroll
    for (int c = 0; c < 3; ++c) {
        bool fg = (t == c);
        f[(b * 3 + c) * VOL + r]     = fg ? 0.0f : INFV;
        f[(6 + b * 3 + c) * VOL + r] = fg ? INFV : 0.0f;
        if (fg) atomicOr(&anyfg[b * 3 + c], 1);
    }
}

// ---------------------------------------------------------------------------
// One separable min-plus pass over all 12 volumes.
// A block owns one 64(line)x64(perp) tile; tile staged in LDS (stride 65 pad).
//   element addr = base + j*lineStride + p*perpStride
//   X pass: lineStride=1,    perpStride=64, sliceStride=4096, transpose=1
//   Y pass: lineStride=64,   perpStride=1,  sliceStride=4096, transpose=0
//   Z pass: lineStride=4096, perpStride=1,  sliceStride=64,   transpose=0
// Load index mapping is chosen so global loads are contiguous in all passes.
// Tile fill uses GLOBAL_LOAD_ASYNC_TO_LDS_B32 when available (ASYNCcnt path).
// ---------------------------------------------------------------------------
__global__ void k_edt_pass(float* __restrict__ f, int lineStride, int perpStride,
                           int sliceStride, int transposeLoad) {
    __shared__ float tile[64 * 65];
    int blk   = blockIdx.x;          // 12 vols * 64 slices = 768 blocks
    int vol   = blk >> 6;
    int slice = blk & 63;
    int base  = vol * VOL + slice * sliceStride;

    for (int i = threadIdx.x; i < 4096; i += 256) {
        int j, p;
        if (transposeLoad) { j = i & 63;  p = i >> 6; }
        else               { j = i >> 6;  p = i & 63; }
        const float* gsrc = f + base + j * lineStride + p * perpStride;
        float*       ldst = tile + j * 65 + p;
#if HAVE_ASYNC_LDS
        __builtin_amdgcn_global_load_async_to_lds_b32(
            (GInt*)(__UINTPTR_TYPE__)gsrc,
            (LInt*)(unsigned)(__UINTPTR_TYPE__)ldst,
            /*offset=*/0, /*cpol=*/0);
#else
        *ldst = *gsrc;
#endif
    }
#if HAVE_ASYNC_LDS
    __builtin_amdgcn_s_wait_asynccnt(0);   // this wave's LDS writes landed
#endif
    __syncthreads();                        // all waves' tile portions visible

    int lane = threadIdx.x & 63;
    int grp  = threadIdx.x >> 6;     // 0..3
#pragma unroll 4
    for (int k = 0; k < 16; ++k) {
        int to, p;
        if (transposeLoad) { to = lane;           p  = grp + 4 * k; }
        else               { to = grp + 4 * k;    p  = lane;        }
        float best = 3.0e38f;
#pragma unroll 8
        for (int j = 0; j < 64; ++j) {
            int d = to - j;
            best = fminf(best, tile[j * 65 + p] + (float)(d * d));
        }
        f[base + to * lineStride + p * perpStride] = best;
    }
}

// ---------------------------------------------------------------------------
// Stage 1 reduce: softmax, absd select, p*d accumulate, WMMA wave reduction.
// 256 blocks x 256 threads = 65536 threads -> exactly 8 voxels/thread (uniform).
// ---------------------------------------------------------------------------
__global__ void k_reduce1(const float* __restrict__ net, const int* __restrict__ tgt,
                          const float* __restrict__ f, const int* __restrict__ anyfg,
                          float* __restrict__ partials) {
    int tid = blockIdx.x * 256 + threadIdx.x;
    float acc = 0.0f;
    for (int s = tid; s < NVOX; s += 65536) {
        int b = s >> 18;
        int r = s & (VOL - 1);
        int t = tgt[s];
        float l0 = net[(b * 3 + 0) * VOL + r];
        float l1 = net[(b * 3 + 1) * VOL + r];
        float l2 = net[(b * 3 + 2) * VOL + r];
        float m  = fmaxf(l0, fmaxf(l1, l2));
        float e0 = expf(l0 - m), e1 = expf(l1 - m), e2 = expf(l2 - m);
        float inv = 1.0f / (e0 + e1 + e2);
        float e[3] = {e0, e1, e2};
#pragma unroll
        for (int c = 0; c < 3; ++c) {
            bool fg = (t == c);
            int  volIdx = (fg ? 6 : 0) + b * 3 + c;   // fg -> dist_in, bg -> dist_out
            float d2 = f[volIdx * VOL + r];
            float dd = sqrtf(d2);
            dd = anyfg[b * 3 + c] ? dd : 0.0f;
            acc += e[c] * inv * dd;
        }
    }
    float s = wave_sum_wmma(acc);                      // uniform point, EXEC all-ones
    __shared__ float wsum[8];
    if ((threadIdx.x & 31) == 0) wsum[threadIdx.x >> 5] = s;
    __syncthreads();
    if (threadIdx.x == 0) {
        float bsum = 0.0f;
#pragma unroll
        for (int w = 0; w < 8; ++w) bsum += wsum[w];
        partials[blockIdx.x] = bsum;
    }
}

// ---------------------------------------------------------------------------
// Stage 2: 256 partials -> scalar mean (deterministic fixed-order combine).
// ---------------------------------------------------------------------------
__global__ void k_reduce2(const float* __restrict__ partials, float* __restrict__ out) {
    float acc = partials[threadIdx.x];                 // exactly 256/256, uniform
    float s = wave_sum_wmma(acc);
    __shared__ float wsum[8];
    if ((threadIdx.x & 31) == 0) wsum[threadIdx.x >> 5] = s;
    __syncthreads();
    if (threadIdx.x == 0) {
        float tsum = 0.0f;
#pragma unroll
        for (int w = 0; w < 8; ++w) tsum += wsum[w];
        out[0] = tsum / 1572864.0f;                    // mean over B*C*Z*Y*X
    }
}

// ---------------------------------------------------------------------------
extern "C" void kernel_launch(void* const* d_in, const int* in_sizes, int n_in,
                              void* d_out, int out_size, void* d_ws, size_t ws_size,
                              hipStream_t stream) {
    (void)in_sizes; (void)n_in; (void)out_size; (void)ws_size;
    const float* net = (const float*)d_in[0];
    const int*   tgt = (const int*)d_in[1];

    float* f        = (float*)d_ws;                                   // 12 * 64^3 f32
    char*  wsb      = (char*)d_ws;
    int*   anyfg    = (int*)(wsb + (size_t)12 * VOL * 4);             // 6 ints
    float* partials = (float*)(wsb + (size_t)12 * VOL * 4 + 64);      // 256 f32

    (void)hipMemsetAsync(anyfg, 0, 6 * sizeof(int), stream);

    k_init<<<NVOX / 256, 256, 0, stream>>>(tgt, f, anyfg);

    // Reference axis order: Z, then Y, then X
    k_edt_pass<<<12 * 64, 256, 0, stream>>>(f, 4096, 1, 64, 0);   // Z
    k_edt_pass<<<12 * 64, 256, 0, stream>>>(f, 64, 1, 4096, 0);   // Y
    k_edt_pass<<<12 * 64, 256, 0, stream>>>(f, 1, 64, 4096, 1);   // X

    k_reduce1<<<256, 256, 0, stream>>>(net, tgt, f, anyfg, partials);
    k_reduce2<<<1, 256, 0, stream>>>(partials, (float*)d_out);
}